// NeighborAttention_21354577396201
// MI455X (gfx1250) — compile-verified
//
#include <hip/hip_runtime.h>
#include <hip/hip_bf16.h>
#include <math.h>

// ---------------------------------------------------------------------------
// NeighborAttention forward for gfx1250 (MI455X, wave32).
// Key fact: query == broadcast of pooled neighbor vector -> q/scores/weights/
// context/out-projection are per-(b,h) vectors. Heavy work is the K and V
// GEMMs, done in full fp32 with V_WMMA_F32_16X16X4_F32.
//
// Sizes: BS=8, SEQ=1024, DIM=1024, H=16, DH=64, NNB=50.
// Workspace layout (floats): xn[8K] qv[8K] ctx[8K] outv[8K] K[8M] V[8M] ~ 67MB
// ---------------------------------------------------------------------------

typedef __attribute__((ext_vector_type(2))) float v2f;
typedef __attribute__((ext_vector_type(8))) float v8f;

#define BS   8
#define SEQ  1024
#define DIM  1024
#define NH   16
#define DH   64
#define NNB  50

// ---------------------------------------------------------------------------
// GEMM: C[M,N] = X[M,K] * W[N,K]^T + bias[N]   (M=8192, N=K=1024)
// Block: 256 threads = 8 waves, block tile 64(M) x 256(N).
// Wave: 32x64 patch = 2x4 WMMA 16x16 tiles, K-loop step 4.
// ---------------------------------------------------------------------------
__global__ __launch_bounds__(256) void gemm_xwT_bias_kernel(
    const float* __restrict__ X, const float* __restrict__ W,
    const float* __restrict__ bias, float* __restrict__ C) {
  const int K = DIM, N = DIM;
  const int lane = threadIdx.x & 31;
  const int wave = threadIdx.x >> 5;
  const int wm = wave & 1;             // 0..1  (M sub-block)
  const int wn = wave >> 1;            // 0..3  (N sub-block)
  const int r0 = blockIdx.x * 64 + wm * 32;   // wave rows r0..r0+31
  const int c0 = blockIdx.y * 256 + wn * 64;  // wave cols c0..c0+63

  const int frow = lane & 15;          // fragment row/col within tile
  const int koff = (lane >> 4) * 2;    // K pair offset (ISA 16x4 layout)

  const float* __restrict__ A0 = X + (size_t)(r0 + frow) * K;       // M tile 0
  const float* __restrict__ A1 = A0 + (size_t)16 * K;               // M tile 1
  const float* __restrict__ B0 = W + (size_t)(c0 + frow) * K;       // N tile j: +16*K*j

  v8f zero = {0.f, 0.f, 0.f, 0.f, 0.f, 0.f, 0.f, 0.f};
  v8f acc[2][4];
#pragma unroll
  for (int i = 0; i < 2; ++i)
#pragma unroll
    for (int j = 0; j < 4; ++j) acc[i][j] = zero;

  for (int k0 = 0; k0 < K; k0 += 4) {
    const int kk = k0 + koff;
    v2f a0 = *(const v2f*)(A0 + kk);
    v2f a1 = *(const v2f*)(A1 + kk);
    v2f b[4];
#pragma unroll
    for (int j = 0; j < 4; ++j) b[j] = *(const v2f*)(B0 + (size_t)16 * j * K + kk);
#pragma unroll
    for (int j = 0; j < 4; ++j) {
      acc[0][j] = __builtin_amdgcn_wmma_f32_16x16x4_f32(
          false, a0, false, b[j], (short)0, acc[0][j], false, false);
      acc[1][j] = __builtin_amdgcn_wmma_f32_16x16x4_f32(
          false, a1, false, b[j], (short)0, acc[1][j], false, false);
    }
  }

  // D layout (32-bit 16x16): N = lane&15, M = vgpr + 8*(lane>>4)
  const int nloc = lane & 15;
  const int mhalf = (lane >> 4) * 8;
#pragma unroll
  for (int i = 0; i < 2; ++i) {
#pragma unroll
    for (int j = 0; j < 4; ++j) {
      const int ncol = c0 + 16 * j + nloc;
      const float bv = bias[ncol];
#pragma unroll
      for (int r = 0; r < 8; ++r) {
        const int m = r0 + i * 16 + mhalf + r;
        C[(size_t)m * N + ncol] = acc[i][j][r] + bv;
      }
    }
  }
}

// ---------------------------------------------------------------------------
// Masked mean-pool of neighbors: xn[b,d] = sum_n x_nbr[b,n,d]*nm[b,n] / sum nm
// ---------------------------------------------------------------------------
__global__ __launch_bounds__(256) void pool_kernel(
    const float* __restrict__ xnbr, const float* __restrict__ nmask,
    float* __restrict__ xn) {
  const int b = blockIdx.x, t = threadIdx.x;
  float denom = 0.f;
  for (int n = 0; n < NNB; ++n) denom += nmask[b * NNB + n];
  const float inv = 1.f / denom;
#pragma unroll
  for (int i = 0; i < 4; ++i) {
    const int d = t + i * 256;
    float s = 0.f;
    for (int n = 0; n < NNB; ++n)
      s += xnbr[((size_t)b * NNB + n) * DIM + d] * nmask[b * NNB + n];
    xn[b * DIM + d] = s * inv;
  }
}

// ---------------------------------------------------------------------------
// vout[b,j] = (sum_d vin[b,d]*W[j,d] + bias[j]) * scale      (8 blocks)
// ---------------------------------------------------------------------------
__global__ __launch_bounds__(256) void matvec_kernel(
    const float* __restrict__ vin, const float* __restrict__ W,
    const float* __restrict__ bias, float scale, float* __restrict__ vout) {
  __shared__ float sx[DIM];
  const int b = blockIdx.x, t = threadIdx.x;
#pragma unroll
  for (int i = 0; i < 4; ++i) sx[t + i * 256] = vin[b * DIM + t + i * 256];
  __syncthreads();
#pragma unroll
  for (int i = 0; i < 4; ++i) {
    const int j = t + i * 256;
    const float* __restrict__ wr = W + (size_t)j * DIM;
    float s = 0.f;
    for (int d = 0; d < DIM; ++d) s += sx[d] * wr[d];
    vout[b * DIM + j] = (s + bias[j]) * scale;
  }
}

// ---------------------------------------------------------------------------
// Attention (one block per (b,h)): scores over 1024 keys, softmax, context.
// ---------------------------------------------------------------------------
__global__ __launch_bounds__(256) void attention_kernel(
    const float* __restrict__ qv, const float* __restrict__ Kmat,
    const float* __restrict__ Vmat, const int* __restrict__ mask,
    float* __restrict__ ctx) {
  __shared__ float sq[DH];
  __shared__ float sw[SEQ];
  __shared__ float red[256];
  __shared__ float s_max, s_inv;
  const int b = blockIdx.x >> 4;
  const int h = blockIdx.x & 15;
  const int t = threadIdx.x;
  if (t < DH) sq[t] = qv[b * DIM + h * DH + t];
  __syncthreads();

  float lmax = -INFINITY;
#pragma unroll
  for (int i = 0; i < 4; ++i) {
    const int kp = t + i * 256;
    const float* __restrict__ kr = Kmat + ((size_t)(b * SEQ + kp)) * DIM + h * DH;
    float s = 0.f;
    for (int d = 0; d < DH; ++d) s += sq[d] * kr[d];
    if (mask[b * SEQ + kp] == 0) s = -INFINITY;
    sw[kp] = s;
    lmax = fmaxf(lmax, s);
  }
  red[t] = lmax;
  __syncthreads();
  for (int sft = 128; sft > 0; sft >>= 1) {
    if (t < sft) red[t] = fmaxf(red[t], red[t + sft]);
    __syncthreads();
  }
  if (t == 0) s_max = red[0];
  __syncthreads();
  const float m = s_max;
  __syncthreads();

  float lsum = 0.f;
#pragma unroll
  for (int i = 0; i < 4; ++i) {
    const int kp = t + i * 256;
    const float e = __expf(sw[kp] - m);
    sw[kp] = e;
    lsum += e;
  }
  red[t] = lsum;
  __syncthreads();
  for (int sft = 128; sft > 0; sft >>= 1) {
    if (t < sft) red[t] += red[t + sft];
    __syncthreads();
  }
  if (t == 0) s_inv = 1.f / red[0];
  __syncthreads();

  // context: dh = t&63, k-chunk = t>>6 (4 chunks of 256 keys)
  const int dh = t & 63;
  const int ch = t >> 6;
  float a = 0.f;
  const float* __restrict__ vbase = Vmat + ((size_t)(b * SEQ)) * DIM + h * DH + dh;
  for (int kp = ch * 256; kp < (ch + 1) * 256; ++kp)
    a += sw[kp] * vbase[(size_t)kp * DIM];
  __syncthreads();
  red[t] = a;
  __syncthreads();
  if (t < DH) {
    const float tot = red[t] + red[64 + t] + red[128 + t] + red[192 + t];
    ctx[b * DIM + h * DH + t] = tot * s_inv;
  }
}

// ---------------------------------------------------------------------------
// y = LayerNorm(outv[b,:] + x[row,:]) * g + beta     (one block per row)
// ---------------------------------------------------------------------------
__global__ __launch_bounds__(256) void resid_ln_kernel(
    const float* __restrict__ x, const float* __restrict__ outv,
    const float* __restrict__ g, const float* __restrict__ beta,
    float* __restrict__ y) {
  __shared__ float red[256];
  __shared__ float s_mu, s_rstd;
  const int row = blockIdx.x;
  const int b = row >> 10;
  const int t = threadIdx.x;
  float h[4];
  float s = 0.f;
#pragma unroll
  for (int i = 0; i < 4; ++i) {
    const int d = t + i * 256;
    h[i] = x[(size_t)row * DIM + d] + outv[b * DIM + d];
    s += h[i];
  }
  red[t] = s;
  __syncthreads();
  for (int sft = 128; sft > 0; sft >>= 1) {
    if (t < sft) red[t] += red[t + sft];
    __syncthreads();
  }
  if (t == 0) s_mu = red[0] * (1.f / DIM);
  __syncthreads();
  const float mu = s_mu;
  float sq = 0.f;
#pragma unroll
  for (int i = 0; i < 4; ++i) {
    const float dv = h[i] - mu;
    sq += dv * dv;
  }
  __syncthreads();
  red[t] = sq;
  __syncthreads();
  for (int sft = 128; sft > 0; sft >>= 1) {
    if (t < sft) red[t] += red[t + sft];
    __syncthreads();
  }
  if (t == 0) s_rstd = rsqrtf(red[0] * (1.f / DIM) + 1e-12f);
  __syncthreads();
  const float rstd = s_rstd;
#pragma unroll
  for (int i = 0; i < 4; ++i) {
    const int d = t + i * 256;
    y[(size_t)row * DIM + d] = (h[i] - mu) * rstd * g[d] + beta[d];
  }
}

// ---------------------------------------------------------------------------
extern "C" void kernel_launch(void* const* d_in, const int* in_sizes, int n_in,
                              void* d_out, int out_size, void* d_ws, size_t ws_size,
                              hipStream_t stream) {
  const float* x      = (const float*)d_in[0];
  const float* xnbr   = (const float*)d_in[1];
  const int*   mask   = (const int*)d_in[2];
  const float* nmask  = (const float*)d_in[3];
  const float* qw     = (const float*)d_in[4];
  const float* qb     = (const float*)d_in[5];
  const float* kw     = (const float*)d_in[6];
  const float* kb     = (const float*)d_in[7];
  const float* vw     = (const float*)d_in[8];
  const float* vb     = (const float*)d_in[9];
  const float* ow     = (const float*)d_in[10];
  const float* ob     = (const float*)d_in[11];
  const float* ln_g   = (const float*)d_in[12];
  const float* ln_b   = (const float*)d_in[13];
  float* out = (float*)d_out;

  float* xn   = (float*)d_ws;                 // BS*DIM
  float* qv   = xn + BS * DIM;                // BS*DIM
  float* ctx  = qv + BS * DIM;                // BS*DIM
  float* outv = ctx + BS * DIM;               // BS*DIM
  float* Kmat = outv + BS * DIM;              // BS*SEQ*DIM
  float* Vmat = Kmat + (size_t)BS * SEQ * DIM;

  pool_kernel<<<BS, 256, 0, stream>>>(xnbr, nmask, xn);
  matvec_kernel<<<BS, 256, 0, stream>>>(xn, qw, qb, 0.125f, qv);  // /sqrt(64)

  dim3 ggrid(BS * SEQ / 64, DIM / 256);
  gemm_xwT_bias_kernel<<<ggrid, 256, 0, stream>>>(x, kw, kb, Kmat);
  gemm_xwT_bias_kernel<<<ggrid, 256, 0, stream>>>(x, vw, vb, Vmat);

  attention_kernel<<<BS * NH, 256, 0, stream>>>(qv, Kmat, Vmat, mask, ctx);
  matvec_kernel<<<BS, 256, 0, stream>>>(ctx, ow, ob, 1.0f, outv);
  resid_ln_kernel<<<BS * SEQ, 256, 0, stream>>>(x, outv, ln_g, ln_b, out);
}